// Gabor_4415226380462
// MI455X (gfx1250) — compile-verified
//
#include <hip/hip_runtime.h>
#include <cstdint>
#include <cstddef>

// Gabor broadcast-multiply: out[g*N + i] = tab[g][i % 81] * x[i]
// N = co*ci*h*w = 21,233,664 (f32), G = 4, h = w = 9.
// Pure streaming kernel: 85 MB in, 340 MB out, ~0.2 FLOP/byte.
// Strategy: async global->LDS staging (CDNA5 ASYNCcnt path), double-buffered
// per wave, b128 everywhere, non-temporal stores for the write-once output.

typedef float f32x4 __attribute__((ext_vector_type(4)));

#define HW   81u
#define WPIX 9u
#define PI_F 3.14159265358979323846f

// One per-lane async 16B copy: global (saddr + 32-bit voffset) -> LDS byte addr.
__device__ __forceinline__ void async_ld_b128(unsigned ldsAddr,
                                              const float* __restrict__ base,
                                              unsigned byteOff) {
    asm volatile("global_load_async_to_lds_b128 %0, %1, %2"
                 :: "v"(ldsAddr), "v"(byteOff), "s"(base)
                 : "memory");
}

__global__ __launch_bounds__(256) void gabor_bcast_kernel(
        const float* __restrict__ x,
        const float* __restrict__ theta,
        const float* __restrict__ lam,
        float* __restrict__ out,
        unsigned N4)                      // number of f32x4 elements in x
{
    __shared__ __align__(16) float tabf[HW][4];   // tab[hw][g]
    __shared__ f32x4 stage[8][2][128];            // [wave][buf][slot] = 32 KB

    // ---- Build the 4x81 Gabor table once per block (negligible cost) ----
    for (unsigned t = threadIdx.x; t < 4u * HW; t += blockDim.x) {
        unsigned g = t / HW;
        unsigned p = t - g * HW;
        float fy = (float)(p / WPIX) - 4.0f;      // (h-1)/2 = 4
        float fx = (float)(p % WPIX) - 4.0f;
        float s, c;
        __sincosf(theta[g], &s, &c);
        float xr = fx * c + fy * s;
        float yr = -fx * s + fy * c;
        float env = __expf(-(xr * xr + yr * yr) / (2.0f * PI_F * PI_F)); // SIGMA = pi
        tabf[p][g] = env * __cosf(2.0f * PI_F * xr * lam[g]);
    }
    __syncthreads();

    const unsigned lane = threadIdx.x & 31u;
    const unsigned wave = threadIdx.x >> 5u;
    const unsigned gw   = blockIdx.x * 8u + wave;
    const unsigned nw   = gridDim.x * 8u;
    const unsigned numTiles = (N4 + 127u) >> 7u;  // 128 f32x4 per tile

    if (gw >= numTiles) return;                   // no barriers past this point

    // ---- Prologue: issue first tile into buffer 0 ----
    {
        unsigned base = gw * 128u + lane;
#pragma unroll
        for (unsigned u = 0; u < 4u; ++u) {
            unsigned i4 = base + u * 32u;
            if (i4 < N4) {
                unsigned lds = (unsigned)(uintptr_t)&stage[wave][0][u * 32u + lane];
                async_ld_b128(lds, x, i4 * 16u);
            }
        }
    }

    unsigned buf = 0u;
    f32x4* __restrict__ outv = (f32x4*)out;

    for (unsigned t = gw; t < numTiles; t += nw) {
        unsigned tn = t + nw;
        // Next tile issues exactly 4 async chunks only if it is fully in range;
        // otherwise drain completely (keeps the wait-count math exact).
        bool nextFull = (tn < numTiles) && (tn * 128u + 128u <= N4);
        if (tn < numTiles) {
            unsigned nbuf = buf ^ 1u;
            unsigned base = tn * 128u + lane;
#pragma unroll
            for (unsigned u = 0; u < 4u; ++u) {
                unsigned i4 = base + u * 32u;
                if (i4 < N4) {
                    unsigned lds = (unsigned)(uintptr_t)&stage[wave][nbuf][u * 32u + lane];
                    async_ld_b128(lds, x, i4 * 16u);
                }
            }
        }
        if (nextFull) {
            // 4 newly in flight; waiting <=4 means the current tile's 4 are done.
            asm volatile("s_wait_asynccnt 4" ::: "memory");
        } else {
            asm volatile("s_wait_asynccnt 0" ::: "memory");
        }

        // ---- Process current tile from LDS ----
        unsigned base = t * 128u + lane;
#pragma unroll
        for (unsigned u = 0; u < 4u; ++u) {
            unsigned i4 = base + u * 32u;
            if (i4 < N4) {
                f32x4 v = stage[wave][buf][u * 32u + lane];   // ds_load_b128
                unsigned e0  = i4 * 4u;
                unsigned hw0 = e0 % HW;
                unsigned hw1 = hw0 + 1u; hw1 = (hw1 == HW) ? 0u : hw1;
                unsigned hw2 = hw1 + 1u; hw2 = (hw2 == HW) ? 0u : hw2;
                unsigned hw3 = hw2 + 1u; hw3 = (hw3 == HW) ? 0u : hw3;
                f32x4 t0 = *(const f32x4*)&tabf[hw0][0];
                f32x4 t1 = *(const f32x4*)&tabf[hw1][0];
                f32x4 t2 = *(const f32x4*)&tabf[hw2][0];
                f32x4 t3 = *(const f32x4*)&tabf[hw3][0];

                f32x4 r0 = (f32x4){v.x * t0.x, v.y * t1.x, v.z * t2.x, v.w * t3.x};
                f32x4 r1 = (f32x4){v.x * t0.y, v.y * t1.y, v.z * t2.y, v.w * t3.y};
                f32x4 r2 = (f32x4){v.x * t0.z, v.y * t1.z, v.z * t2.z, v.w * t3.z};
                f32x4 r3 = (f32x4){v.x * t0.w, v.y * t1.w, v.z * t2.w, v.w * t3.w};

                f32x4* o = outv + i4;
                size_t  s = (size_t)N4;
                __builtin_nontemporal_store(r0, o);
                __builtin_nontemporal_store(r1, o + s);
                __builtin_nontemporal_store(r2, o + 2u * s);
                __builtin_nontemporal_store(r3, o + 3u * s);
            }
        }
        buf ^= 1u;
    }
}

// Defensive scalar fallback for G != 4 (never used for the reference shape).
__global__ void gabor_generic_kernel(const float* __restrict__ x,
                                     const float* __restrict__ theta,
                                     const float* __restrict__ lam,
                                     float* __restrict__ out,
                                     unsigned N, unsigned G) {
    unsigned idx = blockIdx.x * blockDim.x + threadIdx.x;
    unsigned total = G * N;
    if (idx >= total) return;
    unsigned g = idx / N;
    unsigned i = idx - g * N;
    unsigned p = i % HW;
    float fy = (float)(p / WPIX) - 4.0f;
    float fx = (float)(p % WPIX) - 4.0f;
    float s, c;
    __sincosf(theta[g], &s, &c);
    float xr = fx * c + fy * s;
    float yr = -fx * s + fy * c;
    float env = __expf(-(xr * xr + yr * yr) / (2.0f * PI_F * PI_F));
    out[idx] = env * __cosf(2.0f * PI_F * xr * lam[g]) * x[i];
}

extern "C" void kernel_launch(void* const* d_in, const int* in_sizes, int n_in,
                              void* d_out, int out_size, void* d_ws, size_t ws_size,
                              hipStream_t stream) {
    (void)n_in; (void)out_size; (void)d_ws; (void)ws_size;
    const float* x     = (const float*)d_in[0];
    const float* theta = (const float*)d_in[1];
    const float* lam   = (const float*)d_in[2];
    float* out = (float*)d_out;

    unsigned N = (unsigned)in_sizes[0];   // co*ci*h*w = 21,233,664
    unsigned G = (unsigned)in_sizes[1];   // 4

    if (G == 4u && (N & 3u) == 0u) {
        unsigned N4 = N >> 2;
        unsigned numTiles = (N4 + 127u) >> 7u;        // 41,472 tiles of 2 KB
        unsigned blocks = (numTiles + 31u) / 32u;     // ~4 tiles per wave
        if (blocks < 1u) blocks = 1u;
        if (blocks > 8192u) blocks = 8192u;
        gabor_bcast_kernel<<<blocks, 256, 0, stream>>>(x, theta, lam, out, N4);
    } else {
        unsigned total = G * N;
        unsigned blocks = (total + 255u) / 256u;
        gabor_generic_kernel<<<blocks, 256, 0, stream>>>(x, theta, lam, out, N, G);
    }
}